// EfficientShiftFFN_7945689497928
// MI455X (gfx1250) — compile-verified
//
#include <hip/hip_runtime.h>
#include <stdint.h>

// EfficientShiftFFN on gfx1250 (MI455X), streaming copy + tiny per-token decode.
//
// Memory-bound: 1 GiB total traffic -> ~46us floor @ 23.3 TB/s. Compute is
// ~50 ops per 2KB token, so the kernel is built around the CDNA5 async
// global<->LDS DMA path (ASYNCcnt) instead of VGPR staging. One wave32 per
// 512-float token. Lane L owns bytes [16L, 16L+16) of each 512B slice, and
// the 4 async b128 ops step through the tile in contiguous 512B bursts
// (offset 0/512/1024/1536). NT hints: pure stream, no reuse -> keep it out
// of L2.

#define TOKENS_PER_BLOCK 8   // 8 waves * 32 lanes = 256 threads

__global__ __launch_bounds__(256)
void shiftffn_async_kernel(const float* __restrict__ x,
                           float* __restrict__ out,
                           long n_tokens) {
    __shared__ float smem[TOKENS_PER_BLOCK * 512];

    const int lane = threadIdx.x & 31;
    const int wave = threadIdx.x >> 5;
    const long token = (long)blockIdx.x * TOKENS_PER_BLOCK + wave;
    if (token >= n_tokens) return;   // uniform per wave; EXEC stays all-ones

    float* my = &smem[wave * 512];
    const uint64_t tok_byte = (uint64_t)token * 2048ull;

    // LDS byte address for this lane (low 32 bits of the shared-aperture flat
    // address == LDS offset), and per-lane global address (GV mode: 64-bit
    // VGPR address pair). lane*16 -> each async op below is one contiguous
    // 512B burst across the wave.
    const unsigned lds_lane = (unsigned)(uintptr_t)my + (unsigned)lane * 16u;
    const uint64_t gsrc = (uint64_t)(uintptr_t)x   + tok_byte + (uint64_t)lane * 16u;
    const uint64_t gdst = (uint64_t)(uintptr_t)out + tok_byte + (uint64_t)lane * 16u;

    // ---- async DMA: global -> LDS, 2KB tile (4 x contiguous 512B) ----
    asm volatile(
        "global_load_async_to_lds_b128 %0, %1, off offset:0 th:TH_LOAD_NT\n"
        "global_load_async_to_lds_b128 %0, %1, off offset:512 th:TH_LOAD_NT\n"
        "global_load_async_to_lds_b128 %0, %1, off offset:1024 th:TH_LOAD_NT\n"
        "global_load_async_to_lds_b128 %0, %1, off offset:1536 th:TH_LOAD_NT\n"
        :: "v"(lds_lane), "v"(gsrc) : "memory");
    asm volatile("s_wait_asynccnt 0" ::: "memory");

    // ---- per-token decode (reads straight from LDS) ----
    const bool mark  = my[0]  > 0.5f;   // MARK_AX
    const bool fshl  = my[39] > 0.5f;   // OP_SHL
    const bool fshr0 = my[40] > 0.5f;   // OP_SHR

    float vlo = 0.0f, vhi = 0.0f, vsa = 0.0f;
    if (lane < 16) {
        vlo = my[64 + lane];            // ALU_LO window
        vhi = my[80 + lane];            // ALU_HI window
        vsa = my[96 + lane];            // AX_CARRY_LO window
    }
    const unsigned mlo = __builtin_amdgcn_ballot_w32(vlo > 0.5f);
    const unsigned mhi = __builtin_amdgcn_ballot_w32(vhi > 0.5f);
    const unsigned msa = __builtin_amdgcn_ballot_w32(vsa > 0.5f);
    // argmax over booleans == index of first set bit, 0 if none
    const int lo = mlo ? __builtin_ctz(mlo) : 0;
    const int hi = mhi ? __builtin_ctz(mhi) : 0;
    const int sa = msa ? __builtin_ctz(msa) : 0;

    const int  val     = lo + (hi << 4);                 // 8-bit operand
    const bool act_shl = mark && fshl;
    const bool act_shr = mark && fshr0 && !fshl;         // elif priority
    const bool active  = act_shl || act_shr;
    const int  res     = act_shl ? ((val << sa) & 0xFF)
                       : (act_shr ? (val >> sa) : 0);    // already in [0,255]

    if (lane == 0 && active) {
        my[128 + (res & 15)] += 2.0f;   // OUTPUT_LO scatter
        my[144 + (res >> 4)] += 2.0f;   // OUTPUT_HI scatter
    }

    // DS write must be visible to the async store engine before it reads LDS.
    asm volatile("s_wait_dscnt 0" ::: "memory");

    // ---- async DMA: LDS -> global ----
    asm volatile(
        "global_store_async_from_lds_b128 %0, %1, off offset:0 th:TH_STORE_NT\n"
        "global_store_async_from_lds_b128 %0, %1, off offset:512 th:TH_STORE_NT\n"
        "global_store_async_from_lds_b128 %0, %1, off offset:1024 th:TH_STORE_NT\n"
        "global_store_async_from_lds_b128 %0, %1, off offset:1536 th:TH_STORE_NT\n"
        :: "v"(gdst), "v"(lds_lane) : "memory");
    asm volatile("s_wait_asynccnt 0" ::: "memory");
}

extern "C" void kernel_launch(void* const* d_in, const int* in_sizes, int n_in,
                              void* d_out, int out_size, void* d_ws, size_t ws_size,
                              hipStream_t stream) {
    (void)n_in; (void)out_size; (void)d_ws; (void)ws_size;
    const float* x = (const float*)d_in[0];
    float* out = (float*)d_out;
    const long n = (long)in_sizes[0];        // B*SEQ*512
    const long n_tokens = n / 512;           // 262144
    const int blocks = (int)((n_tokens + TOKENS_PER_BLOCK - 1) / TOKENS_PER_BLOCK);
    shiftffn_async_kernel<<<blocks, 256, 0, stream>>>(x, out, n_tokens);
}